// SVDLinear_16578573763428
// MI455X (gfx1250) — compile-verified
//
#include <hip/hip_runtime.h>
#include <math.h>
#include <stdint.h>

typedef __attribute__((ext_vector_type(2))) float v2f;
typedef __attribute__((ext_vector_type(8))) float v8f;

#define Dd 1024
#define Nn 65536

// ws layout (float offsets)
#define WS_E    0          // 1024: exp(log_diagonal)
#define WS_G2   1024       // 1024: gram of q2
#define WS_G1   2048       // 1024: gram of q1
#define WS_S    3072       // 1024: S = V2^T diag(e) V1
#define WS_Z    4096       // 4096: 64x64 combine matrix
#define WS_SUM  8192       // 1  : sum(log_diagonal)
#define WS_UBF  8448       // 65536: U' packed as WMMA B-fragments (phase 1)
#define WS_RBF  (8448+65536) // 65536: R packed as WMMA B-fragments (phase 2, g-major)

// ---------------------------------------------------------------------------
// Kernel 1: grams, S, e = exp(ld), sum(ld). One small reduction per block.
// ---------------------------------------------------------------------------
__global__ __launch_bounds__(256) void k_dots(const float* __restrict__ q1,
                                              const float* __restrict__ q2,
                                              const float* __restrict__ ld,
                                              float* __restrict__ ws) {
  __shared__ float red[256];
  const int b = blockIdx.x, t = threadIdx.x;
  float s = 0.f;
  if (b < 3072) {
    const int task = b >> 10;      // 0: G2, 1: G1, 2: S
    const int idx  = b & 1023;
    const int a = idx >> 5, c = idx & 31;
    if (task == 0) {
      const float* va = q2 + a * Dd; const float* vb = q2 + c * Dd;
      for (int d = t; d < Dd; d += 256) s += va[d] * vb[d];
    } else if (task == 1) {
      const float* va = q1 + a * Dd; const float* vb = q1 + c * Dd;
      for (int d = t; d < Dd; d += 256) s += va[d] * vb[d];
    } else {
      const float* va = q2 + a * Dd; const float* vb = q1 + c * Dd;
      for (int d = t; d < Dd; d += 256) s += va[d] * expf(ld[d]) * vb[d];
    }
  } else {
    for (int d = t; d < Dd; d += 256) { float v = ld[d]; ws[WS_E + d] = expf(v); s += v; }
  }
  red[t] = s; __syncthreads();
  for (int off = 128; off > 0; off >>= 1) { if (t < off) red[t] += red[t + off]; __syncthreads(); }
  if (t == 0) {
    if (b < 3072) ws[1024 + b] = red[0];   // G2 / G1 / S contiguous
    else          ws[WS_SUM]   = red[0];
  }
}

// ---------------------------------------------------------------------------
// Kernel 2: compact-WY T1/T2 (upper triangular, 32x32) and Z (64x64).
//   T[k][k]=beta_k ; T[i][k] = -beta_k * sum_{j=i..k-1} T[i][j]*G[j][k]
//   Z = [[T2, -T2*S*T1],[0, T1]]
// ---------------------------------------------------------------------------
__global__ __launch_bounds__(256) void k_tz(float* __restrict__ ws) {
  __shared__ float sG[1024], sT2[1024], sT1[1024], sS[1024], sW[1024], sM[1024];
  const int t = threadIdx.x;
  for (int i = t; i < 1024; i += 256) { sG[i] = ws[WS_G2 + i]; sT2[i] = 0.f; sT1[i] = 0.f; }
  __syncthreads();
  if (t < 32) sT2[t * 32 + t] = 2.0f / sG[t * 32 + t];
  __syncthreads();
  for (int k = 1; k < 32; k++) {
    if (t < k) {
      float beta = 2.0f / sG[k * 32 + k];
      float a = 0.f;
      for (int j = t; j < k; j++) a += sT2[t * 32 + j] * sG[j * 32 + k];
      sT2[t * 32 + k] = -beta * a;
    }
    __syncthreads();
  }
  for (int i = t; i < 1024; i += 256) sG[i] = ws[WS_G1 + i];
  __syncthreads();
  if (t < 32) sT1[t * 32 + t] = 2.0f / sG[t * 32 + t];
  __syncthreads();
  for (int k = 1; k < 32; k++) {
    if (t < k) {
      float beta = 2.0f / sG[k * 32 + k];
      float a = 0.f;
      for (int j = t; j < k; j++) a += sT1[t * 32 + j] * sG[j * 32 + k];
      sT1[t * 32 + k] = -beta * a;
    }
    __syncthreads();
  }
  for (int i = t; i < 1024; i += 256) sS[i] = ws[WS_S + i];
  __syncthreads();
  for (int i = t; i < 1024; i += 256) {           // W = S * T1
    int r = i >> 5, c = i & 31; float a = 0.f;
    for (int m = 0; m < 32; m++) a += sS[r * 32 + m] * sT1[m * 32 + c];
    sW[i] = a;
  }
  __syncthreads();
  for (int i = t; i < 1024; i += 256) {           // M = T2 * W
    int r = i >> 5, c = i & 31; float a = 0.f;
    for (int m = 0; m < 32; m++) a += sT2[r * 32 + m] * sW[m * 32 + c];
    sM[i] = a;
  }
  __syncthreads();
  for (int i = t; i < 4096; i += 256) {           // assemble Z
    int r = i >> 6, c = i & 63; float v;
    if (r < 32 && c < 32)      v =  sT2[r * 32 + c];
    else if (r < 32)           v = -sM[r * 32 + (c - 32)];
    else if (c >= 32)          v =  sT1[(r - 32) * 32 + (c - 32)];
    else                       v =  0.f;
    ws[WS_Z + i] = v;
  }
}

// ---------------------------------------------------------------------------
// Kernel 3: build U' = [V2 | e*V1] * Z and R = [e*V2 | V1], store both in the
// per-lane WMMA B-fragment layout (ISA 16x16x4 f32: lanes 0-15 hold K={0,1},
// lanes 16-31 hold K={2,3}). Block b handles 16 rows of D.
// U' is packed chunk-major (phase-1 streaming order), R is packed g-major
// (phase-2 streaming order) so both tables are consumed linearly in 8KB stages.
// ---------------------------------------------------------------------------
__global__ __launch_bounds__(256) void k_pack(const float* __restrict__ q1,
                                              const float* __restrict__ q2,
                                              float* __restrict__ ws) {
  __shared__ float sZ[4096];
  __shared__ float sU[16 * 64];
  __shared__ float sUp[16 * 64];
  __shared__ float sR[16 * 64];
  const int b = blockIdx.x, t = threadIdx.x;
  const int d0 = b * 16;
  for (int i = t; i < 4096; i += 256) sZ[i] = ws[WS_Z + i];
  for (int i = t; i < 1024; i += 256) {
    int r = i >> 6, j = i & 63; int d = d0 + r;
    float e = ws[WS_E + d];
    float u, rr;
    if (j < 32) { float q = q2[j * Dd + d];        u = q;     rr = e * q; }
    else        { float q = q1[(j - 32) * Dd + d]; u = e * q; rr = q;     }
    sU[i] = u; sR[i] = rr;
  }
  __syncthreads();
  for (int i = t; i < 1024; i += 256) {           // U' = U * Z
    int r = i >> 6, j = i & 63; float a = 0.f;
    for (int m = 0; m < 64; m++) a += sU[r * 64 + m] * sZ[m * 64 + j];
    sUp[i] = a;
  }
  __syncthreads();
  v2f* Ubf = (v2f*)(ws + WS_UBF);
  v2f* Rbf = (v2f*)(ws + WS_RBF);
  // phase-1 fragments: chunk c=4b+cl over K-dim (=d), tile tt over 64 j-cols
  for (int i = t; i < 512; i += 256) {
    int cl = i >> 7, tt = (i >> 5) & 3, L = i & 31;
    int hh = L >> 4;
    int dloc = 4 * cl + 2 * hh;
    int j = 16 * tt + (L & 15);
    v2f v; v.x = sUp[dloc * 64 + j]; v.y = sUp[(dloc + 1) * 64 + j];
    Ubf[((4 * b + cl) * 4 + tt) * 32 + L] = v;
  }
  // phase-2 fragments, g-major: Rbf[((g*16 + c2)*4 + bt)*32 + L]
  const int g = b >> 2, bt = b & 3;
  for (int i = t; i < 512; i += 256) {
    int c2 = i >> 5, L = i & 31;
    int hh = L >> 4, dl = L & 15;
    int jj = 4 * c2 + 2 * hh;
    v2f v; v.x = sR[dl * 64 + jj]; v.y = sR[dl * 64 + jj + 1];
    Rbf[((g * 16 + c2) * 4 + bt) * 32 + L] = v;
  }
}

// ---------------------------------------------------------------------------
// Main fused kernel.
//  phase 1: C = x*U' (16x64 per wave, v_wmma_f32_16x16x4_f32), B-operands
//           streamed global->LDS with async b128 copies, double buffered.
//  spill:   C tile to padded LDS (stride 68 -> conflict-free b64 reloads).
//  phase 2: out = x.*e - C*R^T + bias, same async-staged B stream.
//  Inner loops are software-pipelined one chunk ahead so ds/load waits
//  overlap with the 4 WMMAs of the previous chunk.
// ---------------------------------------------------------------------------
#define ASYNC_STAGE(ldsa, ga)                                               \
  asm volatile("global_load_async_to_lds_b128 %0, %1, off\n\t"              \
               "global_load_async_to_lds_b128 %0, %1, off offset:16"        \
               :: "v"(ldsa), "v"(ga) : "memory")
#define WAIT_ASYNC(n) asm volatile("s_wait_asynccnt " #n ::: "memory")

__global__ __launch_bounds__(256) void k_main(const float* __restrict__ x,
                                              const float* __restrict__ bias,
                                              const float* __restrict__ ws,
                                              float* __restrict__ out) {
  __shared__ float sP[128 * 68];
  __shared__ __align__(16) float stage[2][2048];   // 2 x 8KB staging buffers
  const int t = threadIdx.x;
  const int w = t >> 5, L = t & 31;
  const int h = L >> 4, l15 = L & 15;
  const int n0 = blockIdx.x * 128 + w * 16;

  const float* e = ws + WS_E;

  // per-thread async-copy addresses: 32B per thread per 8KB stage
  const uint32_t lds0 = (uint32_t)(uintptr_t)&stage[0][t * 8];
  const uint32_t lds1 = (uint32_t)(uintptr_t)&stage[1][t * 8];
  const uint64_t gU = (uint64_t)(uintptr_t)(ws + WS_UBF) + (uint64_t)t * 32;
  const uint64_t gR = (uint64_t)(uintptr_t)(ws + WS_RBF) + (uint64_t)t * 32;
  const v2f* stg0 = (const v2f*)&stage[0][0];
  const v2f* stg1 = (const v2f*)&stage[1][0];

  const v8f z8 = {0.f, 0.f, 0.f, 0.f, 0.f, 0.f, 0.f, 0.f};
  v8f acc[4] = {z8, z8, z8, z8};

  // ---- phase 1: C = x * U' (K = D, 32 stages x 8 chunks of 4) ----
  const float* xa = x + (size_t)(n0 + l15) * Dd + 2 * h;
  ASYNC_STAGE(lds0, gU);
  for (int s = 0; s < 32; ++s) {
    if (s < 31) {
      ASYNC_STAGE((s & 1) ? lds0 : lds1, gU + (uint64_t)(s + 1) * 8192);
      WAIT_ASYNC(0x2);                 // previous stage landed, next in flight
    } else {
      WAIT_ASYNC(0x0);
    }
    __syncthreads();
    const v2f* stg = (s & 1) ? stg1 : stg0;
    // software pipeline: prefetch chunk cl+1 before running chunk cl's WMMAs
    v2f a_cur = *(const v2f*)(xa + 4 * (s * 8));
    v2f b0 = stg[0 * 32 + L];
    v2f b1 = stg[1 * 32 + L];
    v2f b2 = stg[2 * 32 + L];
    v2f b3 = stg[3 * 32 + L];
#pragma unroll
    for (int cl = 0; cl < 8; ++cl) {
      v2f a_nxt, n0v, n1v, n2v, n3v;
      if (cl < 7) {
        a_nxt = *(const v2f*)(xa + 4 * (s * 8 + cl + 1));
        n0v = stg[((cl + 1) * 4 + 0) * 32 + L];
        n1v = stg[((cl + 1) * 4 + 1) * 32 + L];
        n2v = stg[((cl + 1) * 4 + 2) * 32 + L];
        n3v = stg[((cl + 1) * 4 + 3) * 32 + L];
      }
      acc[0] = __builtin_amdgcn_wmma_f32_16x16x4_f32(false, a_cur, false, b0,
                                                     (short)0, acc[0], false, false);
      acc[1] = __builtin_amdgcn_wmma_f32_16x16x4_f32(false, a_cur, false, b1,
                                                     (short)0, acc[1], false, false);
      acc[2] = __builtin_amdgcn_wmma_f32_16x16x4_f32(false, a_cur, false, b2,
                                                     (short)0, acc[2], false, false);
      acc[3] = __builtin_amdgcn_wmma_f32_16x16x4_f32(false, a_cur, false, b3,
                                                     (short)0, acc[3], false, false);
      if (cl < 7) { a_cur = a_nxt; b0 = n0v; b1 = n1v; b2 = n2v; b3 = n3v; }
    }
    __syncthreads();                   // everyone done before buffer reuse
  }

  // ---- spill C tile to LDS (C-layout -> row-major, padded stride 68) ----
#pragma unroll
  for (int tt = 0; tt < 4; ++tt)
#pragma unroll
    for (int r = 0; r < 8; ++r)
      sP[(w * 16 + r + 8 * h) * 68 + 16 * tt + l15] = acc[tt][r];
  __syncthreads();

  // ---- phase 2: out = x.*e - C*R^T + bias (K = 64; 2 stages per g) ----
  const int arow = (w * 16 + l15) * 68 + 2 * h;
  v8f o[4] = {z8, z8, z8, z8};
  ASYNC_STAGE(lds0, gR);
  for (int s = 0; s < 32; ++s) {
    if (s < 31) {
      ASYNC_STAGE((s & 1) ? lds0 : lds1, gR + (uint64_t)(s + 1) * 8192);
      WAIT_ASYNC(0x2);
    } else {
      WAIT_ASYNC(0x0);
    }
    __syncthreads();
    const v2f* stg = (s & 1) ? stg1 : stg0;
    const int cbase = (s & 1) * 8;
    v2f a_cur = *(const v2f*)(&sP[arow + 4 * cbase]);
    v2f b0 = stg[0 * 32 + L];
    v2f b1 = stg[1 * 32 + L];
    v2f b2 = stg[2 * 32 + L];
    v2f b3 = stg[3 * 32 + L];
#pragma unroll
    for (int cl = 0; cl < 8; ++cl) {
      v2f a_nxt, n0v, n1v, n2v, n3v;
      if (cl < 7) {
        a_nxt = *(const v2f*)(&sP[arow + 4 * (cbase + cl + 1)]);
        n0v = stg[((cl + 1) * 4 + 0) * 32 + L];
        n1v = stg[((cl + 1) * 4 + 1) * 32 + L];
        n2v = stg[((cl + 1) * 4 + 2) * 32 + L];
        n3v = stg[((cl + 1) * 4 + 3) * 32 + L];
      }
      o[0] = __builtin_amdgcn_wmma_f32_16x16x4_f32(false, a_cur, false, b0,
                                                   (short)0, o[0], false, false);
      o[1] = __builtin_amdgcn_wmma_f32_16x16x4_f32(false, a_cur, false, b1,
                                                   (short)0, o[1], false, false);
      o[2] = __builtin_amdgcn_wmma_f32_16x16x4_f32(false, a_cur, false, b2,
                                                   (short)0, o[2], false, false);
      o[3] = __builtin_amdgcn_wmma_f32_16x16x4_f32(false, a_cur, false, b3,
                                                   (short)0, o[3], false, false);
      if (cl < 7) { a_cur = a_nxt; b0 = n0v; b1 = n1v; b2 = n2v; b3 = n3v; }
    }
    if (s & 1) {                       // 16 columns-of-64 group finished
      const int g = s >> 1;
#pragma unroll
      for (int tt = 0; tt < 4; ++tt) {
        const int d = (g * 4 + tt) * 16 + l15;
        const float ev = e[d], bv = bias[d];
#pragma unroll
        for (int r = 0; r < 8; ++r) {
          const size_t off = (size_t)(n0 + r + 8 * h) * Dd + d;
          out[off] = x[off] * ev - o[tt][r] + bv;
        }
        o[tt] = z8;
      }
    }
    __syncthreads();
  }
}

// ---------------------------------------------------------------------------
// logabsdet[n] = sum(log_diagonal), broadcast
// ---------------------------------------------------------------------------
__global__ __launch_bounds__(256) void k_ldet(const float* __restrict__ ws,
                                              float* __restrict__ out2) {
  out2[blockIdx.x * 256 + threadIdx.x] = ws[WS_SUM];
}

extern "C" void kernel_launch(void* const* d_in, const int* in_sizes, int n_in,
                              void* d_out, int out_size, void* d_ws, size_t ws_size,
                              hipStream_t stream) {
  const float* x    = (const float*)d_in[0];
  const float* q1   = (const float*)d_in[1];
  const float* q2   = (const float*)d_in[2];
  const float* ld   = (const float*)d_in[3];
  const float* bias = (const float*)d_in[4];
  float* out = (float*)d_out;
  float* ws  = (float*)d_ws;

  k_dots<<<3073, 256, 0, stream>>>(q1, q2, ld, ws);
  k_tz  <<<1,    256, 0, stream>>>(ws);
  k_pack<<<64,   256, 0, stream>>>(q1, q2, ws);
  k_main<<<Nn / 128, 256, 0, stream>>>(x, bias, ws, out);
  k_ldet<<<Nn / 256, 256, 0, stream>>>(ws, out + (size_t)Nn * Dd);
}